// get_model_48284022342259
// MI455X (gfx1250) — compile-verified
//
#include <hip/hip_runtime.h>
#include <math.h>
#include <stdint.h>

// ---------------------------------------------------------------------------
// FreqEnhance pipeline for MI455X (gfx1250).
//   * every linear stage = f32 WMMA GEMM (V_WMMA_F32_16X16X4_F32)
//   * bulk operand staging global->LDS via the Tensor Data Mover
//     (TENSOR_LOAD_TO_LDS, TENSORcnt) and GLOBAL_LOAD_ASYNC_TO_LDS_B128
//     (ASYNCcnt) -- HBM traffic ~1.9 GB vs 23.3 TB/s => ~85 us
//   * DFT bases stored in fragment-pair-interleaved layout [K/2][N][2] so a
//     B-fragment is a single aligned b64 load; bases stay L2-resident
// ---------------------------------------------------------------------------

typedef float v2f __attribute__((ext_vector_type(2)));
typedef float v8f __attribute__((ext_vector_type(8)));
typedef unsigned int u32x4 __attribute__((ext_vector_type(4)));
typedef int i32x4 __attribute__((ext_vector_type(4)));
typedef int i32x8 __attribute__((ext_vector_type(8)));

#define EPS_F 1e-8f

constexpr int Bz = 8, Cc = 64, Hh = 256, Ww = 256;
constexpr int WF  = 129;            // rfft bins along W
constexpr int WFP = 144;            // padded to 9 tiles of 16
constexpr int BC  = Bz * Cc;        // 512 images
constexpr int SPEC_PER = Hh * WFP;  // 36864 floats per (b,c) spectral plane
constexpr int FLAT = Hh * WF;       // 33024 flattened conv length

// ---- WMMA helpers ----------------------------------------------------------
__device__ __forceinline__ v8f wmma4(v2f a, v2f b, v8f c) {
  return __builtin_amdgcn_wmma_f32_16x16x4_f32(
      false, a, false, b, (short)0, c, false, false);
}

__device__ __forceinline__ v8f vzero8() {
  v8f z = {0.f, 0.f, 0.f, 0.f, 0.f, 0.f, 0.f, 0.f};
  return z;
}

// A fragment: rows m0..m0+15, K cols k..k+3, leading dim ld (row-major MxK)
__device__ __forceinline__ v2f loadA(const float* p, int m0, int k, int ld,
                                     int lane) {
  int r = lane & 15;
  int koff = (lane >> 4) << 1;  // 0 or 2
  return *(const v2f*)(p + (size_t)(m0 + r) * ld + k + koff);
}

// B fragment from row-major KxN storage (two strided b32 loads)
__device__ __forceinline__ v2f loadB(const float* p, int n0, int k, int ld,
                                     int lane) {
  int c = lane & 15;
  int koff = (lane >> 4) << 1;
  v2f b;
  b.x = p[(size_t)(k + koff) * ld + n0 + c];
  b.y = p[(size_t)(k + koff + 1) * ld + n0 + c];
  return b;
}

// B fragment from pair-interleaved storage [K/2][N][2]: one aligned b64 load.
// ld2 = 2*N.
__device__ __forceinline__ v2f loadBp(const float* p, int n0, int k, int ld2,
                                      int lane) {
  int c = lane & 15;
  int koff = (lane >> 4) << 1;
  int kp = (k + koff) >> 1;  // k+koff is even
  return *(const v2f*)(p + (size_t)kp * ld2 + (n0 + c) * 2);
}

__device__ __forceinline__ void storeD(float* p, int m0, int n0, int ld,
                                       int lane, v8f d) {
  int c = lane & 15;
  int rb = (lane >> 4) << 3;  // 0 or 8
#pragma unroll
  for (int v = 0; v < 8; ++v) p[(size_t)(m0 + v + rb) * ld + n0 + c] = d[v];
}

// ---- async copy: global -> LDS, 16 bytes per lane (ASYNCcnt) ---------------
__device__ __forceinline__ void async_copy_b128(const float* gsrc,
                                                float* lds_dst) {
  uint32_t lo = (uint32_t)(uintptr_t)lds_dst;  // LDS aperture: low 32 = offset
  asm volatile("global_load_async_to_lds_b128 %0, %1, off"
               :
               : "v"(lo), "v"(gsrc)
               : "memory");
}
__device__ __forceinline__ void async_wait0() {
  asm volatile("s_wait_asynccnt 0x0" ::: "memory");
}

// ---- Tensor Data Mover: 2D tile global -> LDS (TENSORcnt) ------------------
// D# per ISA 8.3/8.4: group0 = {count|flags, lds_addr, gaddr_lo, gaddr_hi|type},
// group1 = {data_size, tensor_dim0/1, tile_dim0/1, tensor_dim0_stride}.
// All dims/strides in data_size (4-byte) units; LDS fill is contiguous.
__device__ __forceinline__ void tdm_load_2d(const float* gaddr,
                                            uint32_t lds_off, uint32_t td0,
                                            uint32_t td1, uint32_t tile0,
                                            uint32_t tile1, uint64_t stride0) {
  uint64_t ga = (uint64_t)(uintptr_t)gaddr;
  u32x4 g0;
  g0.x = 1u;  // count=1, is_restore=0, gather_mode=0
  g0.y = lds_off;
  g0.z = (uint32_t)ga;
  g0.w = ((uint32_t)(ga >> 32) & 0x01FFFFFFu) | (2u << 30);  // type=2 "image"
  i32x8 g1;
  g1[0] = (int)(2u << 16);  // data_size=2 -> 4 bytes; mask/flags 0
  g1[1] = (int)((td0 & 0xFFFFu) << 16);
  g1[2] = (int)(((td0 >> 16) & 0xFFFFu) | ((td1 & 0xFFFFu) << 16));
  g1[3] = (int)(((td1 >> 16) & 0xFFFFu) | ((tile0 & 0xFFFFu) << 16));
  g1[4] = (int)(tile1 & 0xFFFFu);  // tile_dim1 | tile_dim2=0
  g1[5] = (int)(uint32_t)(stride0 & 0xFFFFFFFFull);
  g1[6] = (int)(uint32_t)((stride0 >> 32) & 0xFFFFull);
  g1[7] = 0;
  i32x4 gz = {0, 0, 0, 0};
#if defined(__clang_major__) && __clang_major__ >= 23
  i32x8 gz8 = {0, 0, 0, 0, 0, 0, 0, 0};
  __builtin_amdgcn_tensor_load_to_lds(g0, g1, gz, gz, gz8, 0);
#else
  __builtin_amdgcn_tensor_load_to_lds(g0, g1, gz, gz, 0);
#endif
}
__device__ __forceinline__ void tdm_wait0() {
  __builtin_amdgcn_s_wait_tensorcnt(0);
}

// ---- basis generation ------------------------------------------------------
// BW (pair-interleaved [128][144][2]): cos(2pi*w*kw/256)/16, -sin/16 (0 pad)
// BH (row-major [256][256], symmetric): cos/16, sin/16
// BE (pair-interleaved [72][256][2]):  alpha_k*cos/16, -alpha_k*sin/16 (0 pad)
__global__ __launch_bounds__(256) void init_basis_kernel(
    float* BWc, float* BWs, float* BHc, float* BHs, float* BEc, float* BEs) {
  int i = blockIdx.x * 256 + threadIdx.x;  // exactly 139264 threads
  const float nrm = 0.0625f;               // 1/sqrt(256) per-dimension ortho
  const float step = 6.283185307179586f / 256.0f;
  if (i < 36864) {
    int w = i / WFP, kw = i % WFP;
    float vc = 0.f, vs = 0.f;
    if (kw < WF) {
      float ang = (float)((w * kw) & 255) * step;
      vc = cosf(ang) * nrm;
      vs = -sinf(ang) * nrm;
    }
    int a = (w >> 1) * (2 * WFP) + kw * 2 + (w & 1);  // pair-interleaved
    BWc[a] = vc;
    BWs[a] = vs;
    return;
  }
  i -= 36864;
  if (i < 65536) {
    int a = i >> 8, b = i & 255;
    float ang = (float)((a * b) & 255) * step;
    BHc[i] = cosf(ang) * nrm;
    BHs[i] = sinf(ang) * nrm;
    return;
  }
  i -= 65536;
  if (i < 36864) {
    int k = i >> 8, w = i & 255;
    float vc = 0.f, vs = 0.f;
    if (k < WF) {
      float al = (k == 0 || k == 128) ? 1.f : 2.f;
      float ang = (float)((k * w) & 255) * step;
      vc = al * cosf(ang) * nrm;
      vs = -al * sinf(ang) * nrm;
    }
    int a = (k >> 1) * (2 * Ww) + w * 2 + (k & 1);  // pair-interleaved
    BEc[a] = vc;
    BEs[a] = vs;
  }
}

// ---- stage A: forward row rDFT ---------------------------------------------
// One workgroup per (b,c) image; whole 256 KB image async-staged into LDS
// (GLOBAL_LOAD_ASYNC_TO_LDS_B128); 8 waves x 18 tiles; basis B-frags from L2.
__global__ __launch_bounds__(256) void rowfft_kernel(
    const float* __restrict__ x, float* __restrict__ Or, float* __restrict__ Oi,
    const float* __restrict__ BWc, const float* __restrict__ BWs) {
  extern __shared__ float smem[];  // 65536 floats
  int tid = threadIdx.x, lane = tid & 31, wave = tid >> 5;
  int bc = blockIdx.x;
  const float* xg = x + (size_t)bc * (Hh * Ww);
  for (int e4 = tid; e4 < (Hh * Ww) / 4; e4 += 256)  // 64 iters, uniform
    async_copy_b128(xg + e4 * 4, &smem[e4 * 4]);
  async_wait0();
  __syncthreads();

  float* Orb = Or + (size_t)bc * SPEC_PER;
  float* Oib = Oi + (size_t)bc * SPEC_PER;
#pragma unroll 1
  for (int tt = 0; tt < 18; ++tt) {  // 144 tiles / 8 waves
    int t = wave * 18 + tt;
    int m0 = (t / 9) * 16, n0 = (t % 9) * 16;
    v8f ar = vzero8(), ai = vzero8();
#pragma unroll 2
    for (int k = 0; k < 256; k += 4) {
      v2f a = loadA(smem, m0, k, Ww, lane);           // LDS
      v2f bcf = loadBp(BWc, n0, k, 2 * WFP, lane);    // L2 basis, b64
      v2f bsf = loadBp(BWs, n0, k, 2 * WFP, lane);
      ar = wmma4(a, bcf, ar);
      ai = wmma4(a, bsf, ai);
    }
    storeD(Orb, m0, n0, WFP, lane, ar);
    storeD(Oib, m0, n0, WFP, lane, ai);
  }
}

// ---- stages B/D: column DFT (complex), sgn=+1 forward / -1 inverse ---------
// One workgroup per image; the 295 KB complex spectrum is staged into LDS by
// a single TDM descriptor (2 rows: real plane then imag plane, row stride =
// plane separation). Or = c(x)Xr + sgn*s(x)Xi ; Oi = c(x)Xi - sgn*s(x)Xr
__global__ __launch_bounds__(256) void coldft_kernel(
    const float* __restrict__ Xr, const float* __restrict__ Xi,
    float* __restrict__ Or, float* __restrict__ Oi,
    const float* __restrict__ Bc, const float* __restrict__ Bs, float sgn,
    uint64_t planeStride) {
  extern __shared__ float smem[];  // 2 * 36864 floats
  float* sR = smem;
  float* sI = smem + SPEC_PER;
  int tid = threadIdx.x, lane = tid & 31, wave = tid >> 5;
  int bc = blockIdx.x;
  const float* gR = Xr + (size_t)bc * SPEC_PER;
  if (wave == 0) {
    tdm_load_2d(gR, (uint32_t)(uintptr_t)sR, /*td0=*/SPEC_PER, /*td1=*/2,
                /*tile0=*/SPEC_PER, /*tile1=*/2, /*stride0=*/planeStride);
    tdm_wait0();
  }
  (void)Xi;
  __syncthreads();

  float* Orb = Or + (size_t)bc * SPEC_PER;
  float* Oib = Oi + (size_t)bc * SPEC_PER;
#pragma unroll 1
  for (int tt = 0; tt < 18; ++tt) {
    int t = wave * 18 + tt;
    int m0 = (t / 9) * 16, n0 = (t % 9) * 16;
    v8f ar = vzero8(), ai = vzero8();
#pragma unroll 2
    for (int k = 0; k < 256; k += 4) {
      v2f ac = loadA(Bc, m0, k, Hh, lane);  // L2-resident basis, b64
      v2f as = loadA(Bs, m0, k, Hh, lane);
      as.x *= sgn;
      as.y *= sgn;
      v2f nas;
      nas.x = -as.x;
      nas.y = -as.y;
      v2f br = loadB(sR, n0, k, WFP, lane);  // LDS
      v2f bi = loadB(sI, n0, k, WFP, lane);  // LDS
      ar = wmma4(ac, br, ar);
      ar = wmma4(as, bi, ar);
      ai = wmma4(ac, bi, ai);
      ai = wmma4(nas, br, ai);
    }
    storeD(Orb, m0, n0, WFP, lane, ar);
    storeD(Oib, m0, n0, WFP, lane, ai);
  }
}

// ---- stage C: magnitude gain (flattened K=5 depthwise conv + sigmoid) ------
__global__ __launch_bounds__(256) void weight_kernel(
    const float* __restrict__ inR, const float* __restrict__ inI,
    float* __restrict__ outR, float* __restrict__ outI,
    const float* __restrict__ fw, const float* __restrict__ fb) {
  int idx = blockIdx.x * 256 + threadIdx.x;  // 512*36864 threads exactly
  int bc = idx / SPEC_PER;
  int rem = idx - bc * SPEC_PER;
  int kh = rem / WFP, kw = rem - kh * WFP;
  if (kw >= WF) {  // keep pad columns zero
    outR[idx] = 0.f;
    outI[idx] = 0.f;
    return;
  }
  int base = bc * SPEC_PER;
  int ch = bc & (Cc - 1);
  int f = kh * WF + kw;  // flattened position (row-wrap like the reference)
  float acc = 0.f;
#pragma unroll
  for (int t = 0; t < 5; ++t) {
    int g = f + t - 2;
    if (g >= 0 && g < FLAT) {
      int gh = g / WF, gw = g - gh * WF;
      int si = base + gh * WFP + gw;
      float rr = inR[si], ii = inI[si];
      acc = fmaf(fw[ch * 5 + t], sqrtf(rr * rr + ii * ii), acc);
    }
  }
  float sig = 1.f / (1.f + __expf(-(acc + fb[ch])));
  float wgt = fminf(fmaxf(sig, -1.f), 1.f);
  float r = inR[idx], im = inI[idx];
  float m = sqrtf(r * r + im * im);
  float gmul = 1.f + wgt;
  float orr, oii;
  if (m < EPS_F) {  // reference's safe-phase path for near-zero bins
    float rp = r + EPS_F;
    float sgnv = (rp > 0.f) ? 1.f : ((rp < 0.f) ? -1.f : 0.f);
    float sr = r + EPS_F * sgnv;
    float ph = atan2f(im, sr);
    orr = m * gmul * cosf(ph);
    oii = m * gmul * sinf(ph);
  } else {
    orr = r * gmul;
    oii = im * gmul;
  }
  outR[idx] = orr;
  outI[idx] = oii;
}

// ---- stage C2: 1x1 channel mix in the spectral domain ----------------------
// Workgroup per (plane, batch, 256-col block): TDM stages the strided
// in[64][256] block (64 KB) into LDS; 8 waves x 8 tiles.
__global__ __launch_bounds__(256) void mix_kernel(
    const float* __restrict__ inR, const float* __restrict__ inI,
    float* __restrict__ outR, float* __restrict__ outI,
    const float* __restrict__ mw) {
  extern __shared__ float smem[];  // 64*256 floats
  int tid = threadIdx.x, lane = tid & 31, wave = tid >> 5;
  int blk = blockIdx.x;  // 2 * 8 * 144 blocks
  int nb = blk % 144;
  int t2 = blk / 144;
  int b = t2 % 8;
  int plane = t2 / 8;
  const float* in = plane ? inI : inR;
  float* out = plane ? outI : outR;
  const float* inb = in + (size_t)b * (Cc * SPEC_PER) + nb * 256;
  float* outb = out + (size_t)b * (Cc * SPEC_PER) + nb * 256;

  if (wave == 0) {
    tdm_load_2d(inb, (uint32_t)(uintptr_t)smem, /*td0=*/256, /*td1=*/Cc,
                /*tile0=*/256, /*tile1=*/Cc, /*stride0=*/(uint64_t)SPEC_PER);
    tdm_wait0();
  }
  __syncthreads();

#pragma unroll 1
  for (int tt = 0; tt < 8; ++tt) {  // 64 tiles / 8 waves
    int t = wave * 8 + tt;
    int mt = t / 16, ns = t % 16;
    int m0 = mt * 16, n0 = ns * 16;
    v8f acc = vzero8();
#pragma unroll
    for (int k = 0; k < Cc; k += 4) {
      v2f a = loadA(mw, m0, k, Cc, lane);      // 16 KB weight, L2-hot
      v2f bf = loadB(smem, n0, k, 256, lane);  // LDS
      acc = wmma4(a, bf, acc);
    }
    storeD(outb + n0, m0, 0, SPEC_PER, lane, acc);
  }
}

// ---- stage E: inverse row real synthesis + residual epilogue ---------------
// Workgroup per image; spectrum (A-side) TDM-staged in LDS, basis from L2.
// out[h][w] = x[h][w] + scale * ( Fr (x) BEc + Fi (x) BEs )
__global__ __launch_bounds__(256) void invrow_kernel(
    const float* __restrict__ Fr, const float* __restrict__ Fi,
    const float* __restrict__ x, const float* __restrict__ BEc,
    const float* __restrict__ BEs, const float* __restrict__ scale,
    float* __restrict__ out, uint64_t planeStride) {
  extern __shared__ float smem[];  // 2 * 36864 floats
  float* sR = smem;
  float* sI = smem + SPEC_PER;
  int tid = threadIdx.x, lane = tid & 31, wave = tid >> 5;
  int bc = blockIdx.x;
  const float* gR = Fr + (size_t)bc * SPEC_PER;
  if (wave == 0) {
    tdm_load_2d(gR, (uint32_t)(uintptr_t)sR, /*td0=*/SPEC_PER, /*td1=*/2,
                /*tile0=*/SPEC_PER, /*tile1=*/2, /*stride0=*/planeStride);
    tdm_wait0();
  }
  (void)Fi;
  __syncthreads();

  float s = scale[0];
  const float* xb = x + (size_t)bc * (Hh * Ww);
  float* ob = out + (size_t)bc * (Hh * Ww);
  int c = lane & 15, rb = (lane >> 4) << 3;
#pragma unroll 1
  for (int tt = 0; tt < 32; ++tt) {  // 256 tiles / 8 waves
    int t = wave * 32 + tt;
    int m0 = (t >> 4) * 16, n0 = (t & 15) * 16;
    v8f acc = vzero8();
#pragma unroll 2
    for (int k = 0; k < WFP; k += 4) {
      v2f a_r = loadA(sR, m0, k, WFP, lane);        // LDS
      v2f a_i = loadA(sI, m0, k, WFP, lane);        // LDS
      v2f bcf = loadBp(BEc, n0, k, 2 * Ww, lane);   // L2 basis, b64
      v2f bsf = loadBp(BEs, n0, k, 2 * Ww, lane);
      acc = wmma4(a_r, bcf, acc);
      acc = wmma4(a_i, bsf, acc);
    }
#pragma unroll
    for (int v = 0; v < 8; ++v) {
      int idx = (m0 + v + rb) * Ww + n0 + c;
      ob[idx] = xb[idx] + s * acc[v];
    }
  }
}

// ---------------------------------------------------------------------------
extern "C" void kernel_launch(void* const* d_in, const int* in_sizes, int n_in,
                              void* d_out, int out_size, void* d_ws,
                              size_t ws_size, hipStream_t stream) {
  (void)in_sizes;
  (void)n_in;
  (void)out_size;
  (void)ws_size;
  const float* x  = (const float*)d_in[0];  // [8,64,256,256]
  const float* fw = (const float*)d_in[1];  // [64,1,5]
  const float* fb = (const float*)d_in[2];  // [64]
  const float* mw = (const float*)d_in[3];  // [64,64]
  const float* sc = (const float*)d_in[4];  // [1]
  float* out = (float*)d_out;

  float* w = (float*)d_ws;
  const size_t SPEC = (size_t)BC * SPEC_PER;  // 18,874,368 floats
  float* P0r = w;
  float* P0i = P0r + SPEC;
  float* P1r = P0i + SPEC;
  float* P1i = P1r + SPEC;
  float* BWc = P1i + SPEC;   // 36864 (pair-interleaved)
  float* BWs = BWc + 36864;  // 36864
  float* BHc = BWs + 36864;  // 65536 (row-major, symmetric)
  float* BHs = BHc + 65536;  // 65536
  float* BEc = BHs + 65536;  // 36864 (pair-interleaved)
  float* BEs = BEc + 36864;  // 36864

  const int IMG_LDS  = Hh * Ww * 4;       // 262144 B
  const int SPEC_LDS = 2 * SPEC_PER * 4;  // 294912 B (<= 320 KB WGP LDS)
  const int MIX_LDS  = Cc * 256 * 4;      // 65536 B
  const uint64_t PLANE = (uint64_t)SPEC;  // r->i plane stride in elements

  init_basis_kernel<<<544, 256, 0, stream>>>(BWc, BWs, BHc, BHs, BEc, BEs);
  rowfft_kernel<<<512, 256, IMG_LDS, stream>>>(x, P0r, P0i, BWc, BWs);
  coldft_kernel<<<512, 256, SPEC_LDS, stream>>>(P0r, P0i, P1r, P1i, BHc, BHs,
                                                1.0f, PLANE);
  weight_kernel<<<73728, 256, 0, stream>>>(P1r, P1i, P0r, P0i, fw, fb);
  mix_kernel<<<2304, 256, MIX_LDS, stream>>>(P0r, P0i, P1r, P1i, mw);
  coldft_kernel<<<512, 256, SPEC_LDS, stream>>>(P1r, P1i, P0r, P0i, BHc, BHs,
                                                -1.0f, PLANE);
  invrow_kernel<<<512, 256, SPEC_LDS, stream>>>(P0r, P0i, x, BEc, BEs, sc, out,
                                                PLANE);
}